// GCN_3_47278999995057
// MI455X (gfx1250) — compile-verified
//
#include <hip/hip_runtime.h>
#include <hip/hip_bf16.h>

typedef __attribute__((ext_vector_type(2))) float v2f;
typedef __attribute__((ext_vector_type(8))) float v8f;

static __device__ __forceinline__ int gtid() {
    return blockIdx.x * blockDim.x + threadIdx.x;
}

// ---------------------------------------------------------------------------
// Degree / normalization (computed once, shared by all 3 layers)
// ---------------------------------------------------------------------------
__global__ void init_deg_kernel(float* __restrict__ deg, int n) {
    int i = gtid();
    if (i < n) deg[i] = 1.0f;  // self-loop contributes 1 to every node's degree
}

__global__ void count_deg_kernel(const int* __restrict__ dst, float* __restrict__ deg, int e) {
    int i = gtid();
    if (i < e) atomicAdd(&deg[dst[i]], 1.0f);
}

__global__ void rsqrt_kernel(float* __restrict__ deg, int n) {
    int i = gtid();
    if (i < n) deg[i] = rsqrtf(deg[i]);  // deg >= 1 always (self-loops)
}

// ---------------------------------------------------------------------------
// Linear layer h = x @ W^T via V_WMMA_F32_16X16X4_F32.
// One wave32 computes a 16-row (node) x 16-col tile; K padded to multiple of 4.
// A (16x4 f32): lanes 0-15 hold K=(k0,k0+1), lanes 16-31 hold K=(k0+2,k0+3).
// B (4x16 f32): VGPR0 rows {k0, k0+2}, VGPR1 rows {k0+1, k0+3}, col = lane&15.
// D (16x16 f32): acc[r] = D[row r + (lane>=16 ? 8 : 0)][col = lane&15].
//
// Out-of-range A rows / K slots / B cols are handled with CLAMPED unconditional
// loads + value selects (v_cndmask), never exec-mask branches, so operand setup
// is branch-free and EXEC stays all-ones through the WMMA. The D store uses a
// wave-uniform full-tile fast path (8 unguarded stores) with a guarded
// fallback only for a trailing partial tile.
// ---------------------------------------------------------------------------
template <int K, int NOUT>
__global__ void linear_wmma_kernel(const float* __restrict__ x,
                                   const float* __restrict__ w,   // [NOUT, K]
                                   float* __restrict__ h,         // [n, NOUT]
                                   int n) {
    const int wave = (blockIdx.x * blockDim.x + threadIdx.x) >> 5;
    const int lane = threadIdx.x & 31;
    const int base = wave * 16;
    if (base >= n) return;  // wave-uniform exit; EXEC all-ones afterwards

    const int m     = lane & 15;        // col index (B/D), row index (A)
    const int khalf = (lane >> 4) * 2;  // 0 for lanes 0-15, 2 for lanes 16-31
    const int row   = min(base + m, n - 1);   // clamped: loads always in-bounds
    const int mcl   = (m < NOUT) ? m : 0;     // clamped W column
    const bool colOK = (m < NOUT);

    v8f acc = {};
    constexpr int KC = (K + 3) / 4;
#pragma unroll
    for (int kc = 0; kc < KC; ++kc) {
        const int k0 = kc * 4 + khalf;
        const bool v0 = (k0 + 0) < K;
        const bool v1 = (k0 + 1) < K;
        const int ka0 = v0 ? (k0 + 0) : 0;   // clamped K indices
        const int ka1 = v1 ? (k0 + 1) : 0;
        // Unconditional in-bounds loads, then zero-select padding lanes.
        const float xa0 = x[row * K + ka0];
        const float xa1 = x[row * K + ka1];
        const float wb0 = w[mcl * K + ka0];  // B[k][m] = W[m][k]
        const float wb1 = w[mcl * K + ka1];
        v2f a, b;
        a.x = v0 ? xa0 : 0.0f;
        a.y = v1 ? xa1 : 0.0f;
        b.x = (colOK && v0) ? wb0 : 0.0f;
        b.y = (colOK && v1) ? wb1 : 0.0f;
        acc = __builtin_amdgcn_wmma_f32_16x16x4_f32(
            /*neg_a=*/false, a, /*neg_b=*/false, b,
            /*c_mod=*/(short)0, acc, /*reuse_a=*/false, /*reuse_b=*/false);
    }

    const int mofs = (lane >> 4) * 8;
    if (base + 16 <= n) {
        // Full tile (wave-uniform): unguarded store fast path.
        if (colOK) {
#pragma unroll
            for (int r = 0; r < 8; ++r)
                h[(base + mofs + r) * NOUT + m] = acc[r];
        }
    } else {
        // Trailing partial tile: per-row guarded stores.
        if (colOK) {
#pragma unroll
            for (int r = 0; r < 8; ++r) {
                const int node = base + mofs + r;
                if (node < n) h[node * NOUT + m] = acc[r];
            }
        }
    }
}

// ---------------------------------------------------------------------------
// agg[i] = bias + self-loop term  (fully overwrites dst buffer incl. poison)
// ---------------------------------------------------------------------------
template <int C>
__global__ void init_agg_kernel(const float* __restrict__ h,
                                const float* __restrict__ dinv,
                                const float* __restrict__ bias,
                                float* __restrict__ agg, int n) {
    int i = gtid();
    if (i >= n) return;
    float s = dinv[i];
    s = s * s;  // self-loop norm: dinv[i]*dinv[i]
#pragma unroll
    for (int c = 0; c < C; ++c) agg[i * C + c] = bias[c] + h[i * C + c] * s;
}

// ---------------------------------------------------------------------------
// Edge scatter: agg[dst] += h[src] * dinv[src]*dinv[dst]   (f32 global atomics)
// ---------------------------------------------------------------------------
template <int C>
__global__ void edge_scatter_kernel(const int* __restrict__ src,
                                    const int* __restrict__ dst,
                                    const float* __restrict__ h,
                                    const float* __restrict__ dinv,
                                    float* __restrict__ agg, int e) {
    int i = gtid();
    if (i >= e) return;
    const int s = src[i];
    const int d = dst[i];
    const float wgt = dinv[s] * dinv[d];
    const float* hs = h + (size_t)s * C;
    float* ad = agg + (size_t)d * C;
    if constexpr (C == 4) {
        float4 hv = *reinterpret_cast<const float4*>(hs);
        atomicAdd(ad + 0, hv.x * wgt);
        atomicAdd(ad + 1, hv.y * wgt);
        atomicAdd(ad + 2, hv.z * wgt);
        atomicAdd(ad + 3, hv.w * wgt);
    } else {
        float2 hv = *reinterpret_cast<const float2*>(hs);
        atomicAdd(ad + 0, hv.x * wgt);
        atomicAdd(ad + 1, hv.y * wgt);
    }
}

__global__ void tanh_kernel(const float* __restrict__ in, float* __restrict__ out, int n) {
    int i = gtid();
    if (i < n) out[i] = tanhf(in[i]);
}

// ---------------------------------------------------------------------------
// Launcher
// ---------------------------------------------------------------------------
static inline int cdiv(int a, int b) { return (a + b - 1) / b; }

extern "C" void kernel_launch(void* const* d_in, const int* in_sizes, int n_in,
                              void* d_out, int out_size, void* d_ws, size_t ws_size,
                              hipStream_t stream) {
    const float* x  = (const float*)d_in[0];
    const int*   ei = (const int*)d_in[1];   // [2, E] int32 (JAX default x64-off)
    const float* W1 = (const float*)d_in[2];
    const float* b1 = (const float*)d_in[3];
    const float* W2 = (const float*)d_in[4];
    const float* b2 = (const float*)d_in[5];
    const float* W3 = (const float*)d_in[6];
    const float* b3 = (const float*)d_in[7];
    float* out = (float*)d_out;

    const int n = in_sizes[0] / 9;      // 100000 nodes
    const int e = in_sizes[1] / 2;      // 3200000 edges
    const int* src = ei;
    const int* dst = ei + e;

    // Workspace carve (256B aligned): dinv[n], bufA[n*4], bufB[n*4], bufC[n*4]
    char* ws = (char*)d_ws;
    size_t off = 0;
    auto carve = [&](size_t bytes) {
        void* p = ws + off;
        off += (bytes + 255) & ~(size_t)255;
        return p;
    };
    float* dinv = (float*)carve((size_t)n * sizeof(float));
    float* bufA = (float*)carve((size_t)n * 4 * sizeof(float));  // h (pre-agg)
    float* bufB = (float*)carve((size_t)n * 4 * sizeof(float));  // agg
    float* bufC = (float*)carve((size_t)n * 4 * sizeof(float));  // activated
    (void)ws_size;

    const int B = 256;
    const int gN = cdiv(n, B);
    const int gE = cdiv(e, B);
    const int gW = cdiv(cdiv(n, 16) * 32, B);  // WMMA: 32 threads per 16 nodes

    // --- shared normalization ---
    init_deg_kernel<<<gN, B, 0, stream>>>(dinv, n);
    count_deg_kernel<<<gE, B, 0, stream>>>(dst, dinv, e);
    rsqrt_kernel<<<gN, B, 0, stream>>>(dinv, n);

    // --- layer 1: 9 -> 4, tanh ---
    linear_wmma_kernel<9, 4><<<gW, B, 0, stream>>>(x, W1, bufA, n);
    init_agg_kernel<4><<<gN, B, 0, stream>>>(bufA, dinv, b1, bufB, n);
    edge_scatter_kernel<4><<<gE, B, 0, stream>>>(src, dst, bufA, dinv, bufB, e);
    tanh_kernel<<<cdiv(n * 4, B), B, 0, stream>>>(bufB, bufC, n * 4);

    // --- layer 2: 4 -> 4, tanh ---
    linear_wmma_kernel<4, 4><<<gW, B, 0, stream>>>(bufC, W2, bufA, n);
    init_agg_kernel<4><<<gN, B, 0, stream>>>(bufA, dinv, b2, bufB, n);
    edge_scatter_kernel<4><<<gE, B, 0, stream>>>(src, dst, bufA, dinv, bufB, e);
    tanh_kernel<<<cdiv(n * 4, B), B, 0, stream>>>(bufB, bufC, n * 4);

    // --- layer 3: 4 -> 2, no activation; aggregate straight into d_out ---
    linear_wmma_kernel<4, 2><<<gW, B, 0, stream>>>(bufC, W3, bufA, n);
    init_agg_kernel<2><<<gN, B, 0, stream>>>(bufA, dinv, b3, out, n);
    edge_scatter_kernel<2><<<gE, B, 0, stream>>>(src, dst, bufA, dinv, out, e);
}